// PatchAggregator_14800457302125
// MI455X (gfx1250) — compile-verified
//
#include <hip/hip_runtime.h>
#include <hip/hip_bf16.h>

typedef __attribute__((ext_vector_type(2))) float v2f;
typedef __attribute__((ext_vector_type(8))) float v8f;

#define S_LEN   4096
#define D_DIM   1024
#define NPATCH  8          // MAX_N_LATENT
#define NSEG    9          // segments 0..8 (8 is discarded)

// ---------------------------------------------------------------------------
// Kernel 1: per-batch segment boundaries.
// Segment p (p=1..8) begins at (index of p-th one) + 1; segment 0 begins at 0.
// ofs[b*10 + p] for p=0..9, with ofs[...,9] = S. One block per batch.
// ---------------------------------------------------------------------------
__global__ void seg_bounds_kernel(const int* __restrict__ mask,
                                  int* __restrict__ ofs) {
  const int b   = blockIdx.x;
  const int tid = threadIdx.x;        // 256 threads, 16 elems each
  __shared__ int psum[256];

  const int base = b * S_LEN + tid * 16;
  int m[16];
  int c = 0;
#pragma unroll
  for (int j = 0; j < 16; ++j) { m[j] = mask[base + j]; c += m[j]; }
  psum[tid] = c;
  __syncthreads();

  // Hillis-Steele inclusive scan over the 256 per-thread counts.
#pragma unroll
  for (int off = 1; off < 256; off <<= 1) {
    int v   = psum[tid];
    int add = (tid >= off) ? psum[tid - off] : 0;
    __syncthreads();
    psum[tid] = v + add;
    __syncthreads();
  }
  const int excl = (tid == 0) ? 0 : psum[tid - 1];

  // Defaults: ofs[0]=0, ofs[1..9]=S (segment empty if the p-th one is absent).
  if (tid < 10) ofs[b * 10 + tid] = (tid == 0) ? 0 : S_LEN;
  __syncthreads();

  int r = excl;                       // rank (= #ones strictly before element)
#pragma unroll
  for (int j = 0; j < 16; ++j) {
    if (m[j]) {
      if (r < NPATCH) ofs[b * 10 + (r + 1)] = tid * 16 + j + 1;
      ++r;
    }
  }
}

// ---------------------------------------------------------------------------
// Kernel 2: segment means for p=0..7 only (segment 8 is discarded by the
// reference: patches = mean[:, :8]). One block per (b,p); thread d sums its
// column over the contiguous row range [ofs[p], ofs[p+1]).
// ---------------------------------------------------------------------------
__global__ void seg_mean_kernel(const float* __restrict__ x,
                                const int* __restrict__ ofs,
                                float* __restrict__ patches) {
  const int p = blockIdx.x & 7;
  const int b = blockIdx.x >> 3;
  const int d = threadIdx.x;          // 1024 threads, one per d

  const int s0 = ofs[b * 10 + p];
  const int s1 = ofs[b * 10 + p + 1];

  const float* src = x + (size_t)b * S_LEN * D_DIM + d;
  float acc = 0.0f;
  for (int s = s0; s < s1; ++s) acc += src[(size_t)s * D_DIM];

  const int cnt = s1 - s0;
  const float denom = (float)(cnt > 0 ? cnt : 1);
  patches[(size_t)(b * NPATCH + p) * D_DIM + d] = acc / denom;
}

// ---------------------------------------------------------------------------
// Kernel 3: out(64,1024) = patches(64,1024) @ W^T + bias, fp32 WMMA 16x16x4.
// One wave per 16x16 output tile. A[M=row on lanes 0-15, K in VGPRs],
// B[K][N]: N on lanes, K in VGPRs; in both, lanes 0-15 hold K={k,k+1} and
// lanes 16-31 hold K={k+2,k+3} -> a single contiguous b64 load per operand.
// C/D: VGPR r -> rows m0+r (lanes 0-15) and m0+8+r (lanes 16-31), N = lane%16.
// ---------------------------------------------------------------------------
__global__ void patch_gemm_wmma(const float* __restrict__ P,
                                const float* __restrict__ Wm,
                                const float* __restrict__ bias,
                                float* __restrict__ out) {
  const int wave = blockIdx.x * (blockDim.x >> 5) + (threadIdx.x >> 5);
  const int lane = threadIdx.x & 31;
  const int mt   = wave >> 6;         // 0..3   (64 rows / 16)
  const int nt   = wave & 63;         // 0..63  (1024 cols / 16)
  const int m0   = mt * 16;
  const int n0   = nt * 16;
  const int half = lane >> 4;         // 0: K pair {0,1}, 1: K pair {2,3}
  const int lr   = lane & 15;

  const float* Arow = P  + (size_t)(m0 + lr) * D_DIM + half * 2;
  const float* Brow = Wm + (size_t)(n0 + lr) * D_DIM + half * 2;  // B[k][n]=W[n][k]

  v8f c = {};
#pragma unroll 8
  for (int k = 0; k < D_DIM; k += 4) {
    v2f a  = *(const v2f*)(Arow + k);
    v2f bb = *(const v2f*)(Brow + k);
    c = __builtin_amdgcn_wmma_f32_16x16x4_f32(
        /*neg_a=*/false, a, /*neg_b=*/false, bb,
        /*c_mod=*/(short)0, c, /*reuse_a=*/false, /*reuse_b=*/false);
  }

  const float bv = bias[n0 + lr];
  float* ocol = out + n0 + lr;
#pragma unroll
  for (int r = 0; r < 8; ++r) {
    ocol[(size_t)(m0 + half * 8 + r) * D_DIM] = c[r] + bv;
  }
}

// ---------------------------------------------------------------------------
extern "C" void kernel_launch(void* const* d_in, const int* in_sizes, int n_in,
                              void* d_out, int out_size, void* d_ws, size_t ws_size,
                              hipStream_t stream) {
  const float* latent = (const float*)d_in[0];   // (8,4096,1024) f32
  const int*   mask   = (const int*)  d_in[1];   // (8,4096,1) i32
  const float* W      = (const float*)d_in[2];   // (1024,1024) f32
  const float* bias   = (const float*)d_in[3];   // (1024,) f32
  float*       out    = (float*)d_out;           // (8,8,1024) f32

  int*   ofs     = (int*)d_ws;                           // 8*10 ints
  float* patches = (float*)((char*)d_ws + 1024);         // 8*8*1024 f32

  seg_bounds_kernel<<<8, 256, 0, stream>>>(mask, ofs);
  seg_mean_kernel<<<64, 1024, 0, stream>>>(latent, ofs, patches);
  patch_gemm_wmma<<<64, 128, 0, stream>>>(patches, W, bias, out);
}